// TanhRT_41085657153638
// MI455X (gfx1250) — compile-verified
//
#include <hip/hip_runtime.h>

#define EPSILON 0.1f

typedef __attribute__((ext_vector_type(2))) float v2f;
typedef __attribute__((ext_vector_type(4))) float v4f;
typedef __attribute__((ext_vector_type(8))) float v8f;
typedef __attribute__((ext_vector_type(4))) int   v4i;

__device__ __constant__ float c_eta_fault[12][4] = {
  { 0.99997f, -1.0f,      0.0f,      9.1182e-18f},
  {-0.55856f, -1.0f,      0.0f,     -3.7472e-10f},
  {-16.919f,  14.562f,    0.086479f, 82.578f},
  { 1.0f,     -1.0f,      0.0f,      4.4982e-16f},
  { 1.6839f,  -2.2404f,  -2.5173f,   1.1147f},
  {-0.55856f, -1.0f,      0.0f,     -3.7472e-10f},
  { 0.99997f, -1.0f,      0.0f,      9.1182e-18f},
  { 0.23198f, -0.78334f, -0.41194f,  3.4699f},
  { 1.0f,     -1.0f,      0.0f,      4.4982e-16f},
  { 0.99997f, -1.0f,      0.0f,      9.1182e-18f},
  {-0.24548f, -0.02379f,  0.81499f,  1.5337f},
  {-1.0f,     -1.0f,      0.0f,      4.407e-16f}
};

__device__ __forceinline__ float fast_tanhf(float x) {
#if __has_builtin(__builtin_amdgcn_tanhf)
  return __builtin_amdgcn_tanhf(x);          // gfx1250 v_tanh_f32
#else
  // tanh(x) = 1 - 2/(exp2(2*log2e*x) + 1); saturates correctly at +/-inf
  float t = __builtin_amdgcn_exp2f(x * 2.8853900817779268f);
  return 1.0f - 2.0f * __builtin_amdgcn_rcpf(t + 1.0f);
#endif
}

// ---------------------------------------------------------------------------
// Kernel 1: tiny MLP producing eta (64 x 4) via V_WMMA_F32_16X16X4_F32.
// One workgroup, 512 threads = 16 wave32s.
// ---------------------------------------------------------------------------
__global__ __launch_bounds__(512) void eta_mlp_kernel(
    const float* __restrict__ rt_,  const float* __restrict__ noise,
    const float* __restrict__ X_max, const float* __restrict__ X_min,
    const float* __restrict__ Y_max, const float* __restrict__ Y_min,
    const float* __restrict__ W1,   const float* __restrict__ b1,
    const float* __restrict__ W2,   const float* __restrict__ b2,
    float* __restrict__ eta_out)
{
  __shared__ float sRT[9], sXmin[9], sInvR[9];
  __shared__ float sXn[64 * 12];   // xn rows, K padded 9 -> 12
  __shared__ float sH[64 * 68];    // h (64x64), row stride 68 to dodge bank conflicts

  const int tid = threadIdx.x;

  // --- per-feature constants ---
  if (tid < 9) {
    float xmin = X_min[tid], xmax = X_max[tid];
    float rtn = 0.0f;
    if (tid < 6) rtn = 1.0f / (1.0f + expf(-rt_[tid]));
    sRT[tid]   = rtn * (xmax - xmin) + xmin;
    sXmin[tid] = xmin;
    sInvR[tid] = 1.0f / (xmax - xmin);
  }
  __syncthreads();

  // --- per-row feature construction: xn (64 x 9, zero-padded to K=12) ---
  if (tid < 64) {
    const int n = tid;
    float rtv[6];
    #pragma unroll
    for (int k = 0; k < 6; ++k) {
      float nf = (noise[n * 9 + k] * 2.0f - 1.0f) * EPSILON + 1.0f;
      rtv[k] = sRT[k] * nf;
    }
    float ext[9];
    ext[0] = rtv[0]; ext[1] = rtv[1]; ext[2] = rtv[2];
    ext[3] = rtv[3]; ext[4] = rtv[4]; ext[5] = rtv[5];
    ext[6] = rtv[1] / rtv[0];
    ext[7] = rtv[3] / rtv[2];
    ext[8] = rtv[5] / rtv[4];
    #pragma unroll
    for (int k = 0; k < 9; ++k)
      sXn[n * 12 + k] = (ext[k] - sXmin[k]) * sInvR[k];
    sXn[n * 12 + 9]  = 0.0f;
    sXn[n * 12 + 10] = 0.0f;
    sXn[n * 12 + 11] = 0.0f;
  }
  __syncthreads();

  const int wave = tid >> 5;
  const int lane = tid & 31;
  const int l16  = lane & 15;
  const int kofs = (lane >= 16) ? 2 : 0;   // A/B lane-half K offset
  const int rofs = (lane >= 16) ? 8 : 0;   // C/D lane-half row offset

  // --- GEMM1: h = relu(xn @ W1 + b1), 64x64, K=12 ; one 16x16 tile per wave ---
  {
    const int Mbase = (wave >> 2) * 16;
    const int Nbase = (wave & 3) * 16;
    v8f acc = {};
    #pragma unroll
    for (int ks = 0; ks < 12; ks += 4) {
      const int ka = ks + kofs;
      v2f a, b;
      a.x = sXn[(Mbase + l16) * 12 + ka];
      a.y = sXn[(Mbase + l16) * 12 + ka + 1];
      const int col = Nbase + l16;
      b.x = (ka     < 9) ? W1[ka * 64 + col]       : 0.0f;
      b.y = (ka + 1 < 9) ? W1[(ka + 1) * 64 + col] : 0.0f;
      acc = __builtin_amdgcn_wmma_f32_16x16x4_f32(
          false, a, false, b, (short)0, acc, false, false);
    }
    const float bias = b1[Nbase + l16];
    #pragma unroll
    for (int g = 0; g < 8; ++g) {
      const int row = Mbase + rofs + g;
      float v = acc[g] + bias;
      sH[row * 68 + (Nbase + l16)] = v > 0.0f ? v : 0.0f;
    }
  }
  __syncthreads();

  // --- GEMM2: eta = sigmoid(h @ W2 + b2)*(Ymax-Ymin)+Ymin, 64x4 (N padded 16) ---
  if (wave < 4) {
    const int Mbase = wave * 16;
    v8f acc = {};
    #pragma unroll
    for (int ks = 0; ks < 64; ks += 4) {
      const int ka = ks + kofs;
      v2f a, b;
      a.x = sH[(Mbase + l16) * 68 + ka];
      a.y = sH[(Mbase + l16) * 68 + ka + 1];
      b.x = (l16 < 4) ? W2[ka * 4 + l16]       : 0.0f;
      b.y = (l16 < 4) ? W2[(ka + 1) * 4 + l16] : 0.0f;
      acc = __builtin_amdgcn_wmma_f32_16x16x4_f32(
          false, a, false, b, (short)0, acc, false, false);
    }
    if (l16 < 4) {
      const float bb = b2[l16], ymin = Y_min[l16], ymax = Y_max[l16];
      #pragma unroll
      for (int g = 0; g < 8; ++g) {
        const int row = Mbase + rofs + g;
        float s = 1.0f / (1.0f + expf(-(acc[g] + bb)));
        eta_out[row * 4 + l16] = s * (ymax - ymin) + ymin;
      }
    }
  }
}

// ---------------------------------------------------------------------------
// Kernel 2: bandwidth-bound elementwise pass over z (4 x 64 x 256 x 512).
// One float4 (B128) of z per thread; NT B128 store of the result.
// ---------------------------------------------------------------------------
__global__ __launch_bounds__(256) void tanh_apply_kernel(
    const float* __restrict__ z, const int* __restrict__ mask,
    const float* __restrict__ eta, float* __restrict__ out)
{
  const long long tid = (long long)blockIdx.x * blockDim.x + threadIdx.x;
  const long long e   = tid * 4;                  // flat element index
  const int c  = (int)(e & 511);                  // innermost dim C = 512
  const long long rem = e >> 9;                   // (m*64 + n)*256 + b
  const int mn = (int)(rem >> 8);
  const int n  = mn & 63;
  const int m  = mn >> 6;

  const v4f zv   = *(const v4f*)(z + e);
  const v4i mk   = *(const v4i*)(mask + m * 512 + c);
  const v4f etaN = *(const v4f*)(eta + n * 4);

  const float zr[4]  = {zv.x, zv.y, zv.z, zv.w};
  const int   mks[4] = {mk.x, mk.y, mk.z, mk.w};

  float res[4];
  #pragma unroll
  for (int j = 0; j < 4; ++j) {
    float e0, e1, e2, e3;
    if (mks[j] == 0) {
      e0 = etaN.x; e1 = etaN.y; e2 = etaN.z; e3 = etaN.w;
    } else {
      const float* f = c_eta_fault[mks[j] - 1];
      e0 = f[0]; e1 = f[1]; e2 = f[2]; e3 = f[3];
    }
    res[j] = e0 + e1 * fast_tanhf((zr[j] - e2) * e3);
  }
  v4f r;
  r.x = res[0]; r.y = res[1]; r.z = res[2]; r.w = res[3];
  __builtin_nontemporal_store(r, (v4f*)(out + e));   // keep z L2-resident
}

extern "C" void kernel_launch(void* const* d_in, const int* in_sizes, int n_in,
                              void* d_out, int out_size, void* d_ws, size_t ws_size,
                              hipStream_t stream) {
  const float* z     = (const float*)d_in[0];
  const float* rt_   = (const float*)d_in[1];
  const float* noise = (const float*)d_in[2];
  const float* X_max = (const float*)d_in[3];
  const float* X_min = (const float*)d_in[4];
  const float* Y_max = (const float*)d_in[5];
  const float* Y_min = (const float*)d_in[6];
  const float* W1    = (const float*)d_in[7];
  const float* b1    = (const float*)d_in[8];
  const float* W2    = (const float*)d_in[9];
  const float* b2    = (const float*)d_in[10];
  const int*   mask  = (const int*)d_in[11];
  float* out = (float*)d_out;
  float* eta = (float*)d_ws;                      // 64*4 floats of scratch

  eta_mlp_kernel<<<1, 512, 0, stream>>>(rt_, noise, X_max, X_min, Y_max, Y_min,
                                        W1, b1, W2, b2, eta);

  const long long nthreads = (long long)out_size / 4;   // 8,388,608
  const int threads = 256;
  const int blocks  = (int)(nthreads / threads);        // 32,768 (exact)
  tanh_apply_kernel<<<blocks, threads, 0, stream>>>(z, mask, eta, out);
}